// HL_35098472743364
// MI455X (gfx1250) — compile-verified
//
#include <hip/hip_runtime.h>
#include <hip/hip_bf16.h>

typedef __attribute__((ext_vector_type(2))) float v2f;
typedef __attribute__((ext_vector_type(8))) float v8f;

// ---------------------------------------------------------------------------
// Phase 1: lattice recurrence over digit-sum levels (batch independent).
// Single workgroup, lattice lives in LDS, one barrier per level.
// node i (0..4095): digits are 2-bit fields; stride for dim with shift sh is 1<<sh.
// ---------------------------------------------------------------------------
__global__ __launch_bounds__(1024) void hl_lattice_kernel(const float* __restrict__ b,
                                                          float* __restrict__ lat) {
    __shared__ float s[4096];
    const int t = threadIdx.x;
    for (int lvl = 0; lvl <= 18; ++lvl) {
        #pragma unroll
        for (int k = 0; k < 4; ++k) {
            const int i = t + k * 1024;
            const int dsum = (i & 3) + ((i >> 2) & 3) + ((i >> 4) & 3) +
                             ((i >> 6) & 3) + ((i >> 8) & 3) + ((i >> 10) & 3);
            if (dsum == lvl) {
                float lb = 0.0f;  // reference floors the max at 0.0 via where(valid,...,0.0)
                #pragma unroll
                for (int sh = 0; sh <= 10; sh += 2) {
                    if ((i >> sh) & 3) lb = fmaxf(lb, s[i - (1 << sh)]);
                }
                const float w = 1.0f / (1.0f + expf(-b[i]));   // sigmoid
                s[i] = lb + w * (1.0f - lb);
            }
        }
        __syncthreads();
    }
    #pragma unroll
    for (int k = 0; k < 4; ++k) lat[t + k * 1024] = s[t + k * 1024];
}

// ---------------------------------------------------------------------------
// Phase 2: 6-D multilinear interpolation, 64 corners per row.
// Per wave: 16 batch rows, corners as K. 16x V_WMMA_F32_16X16X4_F32 accumulate
// the 64-term dot products in exact f32; result read off the D diagonal.
// A 16x4 f32 layout: lanes 0-15 M=0..15 (K=0/1 in VGPR0/1), lanes 16-31 K=2/3.
// => lane L and L+16 handle the same batch row with different corner pairs.
// ---------------------------------------------------------------------------
__global__ __launch_bounds__(256) void hl_interp_kernel(const float* __restrict__ x,
                                                        const float* __restrict__ lat,
                                                        float* __restrict__ out) {
    __shared__ float sl[4096];          // lattice (16 KB)
    __shared__ float sd[8 * 32 * 8];    // per-wave diagonal scratch (8 KB)

    const int t = threadIdx.x;
    const int wave = t >> 5;
    const int lane = t & 31;
    const int half = lane >> 4;     // selects corner pair offset (K split)
    const int m    = lane & 15;     // batch row within the wave's 16-row tile
    const int row0 = blockIdx.x * 128 + wave * 16;
    const int row  = row0 + m;

    __builtin_prefetch(&x[row * 6], 0, 1);   // global_prefetch_b8

    // Stage lattice into LDS: 4 float4 per thread.
    const float4* lat4 = (const float4*)lat;
    float4* sl4 = (float4*)sl;
    #pragma unroll
    for (int k = 0; k < 4; ++k) sl4[t + k * 256] = lat4[t + k * 256];
    __syncthreads();

    // Per-dimension fractions and base corner index. STRIDES[j] = 4^(5-j) = 1024 >> 2j.
    float frac[6], omf[6];
    int cornerBase = 0;
    #pragma unroll
    for (int j = 0; j < 6; ++j) {
        float xv = x[row * 6 + j];
        xv = fminf(fmaxf(xv, 0.0f), 1.0f);
        const float coord = xv * 3.0f;            // (size-1)
        float fl = floorf(coord);
        fl = fminf(fl, 2.0f);                     // clip to size-2
        frac[j] = coord - fl;
        omf[j]  = 1.0f - frac[j];
        cornerBase += (int)fl * (1024 >> (2 * j));
    }

    auto wcorner = [&](int c) -> float {
        float w = 1.0f;
        #pragma unroll
        for (int j = 0; j < 6; ++j) w *= ((c >> j) & 1) ? frac[j] : omf[j];
        return w;
    };
    auto vcorner = [&](int c) -> float {
        int off = 0;
        #pragma unroll
        for (int j = 0; j < 6; ++j)
            if ((c >> j) & 1) off += (1024 >> (2 * j));
        return sl[cornerBase + off];
    };

    // Accumulate 64 corners as 16 chunks of K=4 in exact f32 WMMA.
    v8f acc = {};
    #pragma unroll
    for (int kc = 0; kc < 16; ++kc) {
        const int c0 = kc * 4 + half * 2;   // this lane's K-pair within the chunk
        v2f a, bm;
        a[0]  = wcorner(c0);
        a[1]  = wcorner(c0 + 1);
        bm[0] = vcorner(c0);
        bm[1] = vcorner(c0 + 1);
        acc = __builtin_amdgcn_wmma_f32_16x16x4_f32(
            /*neg_a=*/false, a, /*neg_b=*/false, bm,
            /*c_mod=*/(short)0, acc, /*reuse_a=*/false, /*reuse_b=*/false);
    }

    // Diagonal extraction: D layout -> VGPR r: lanes 0-15 M=r, lanes 16-31 M=r+8; N=lane%16.
    float* sc = sd + wave * 256;
    #pragma unroll
    for (int r = 0; r < 8; ++r) sc[lane * 8 + r] = acc[r];
    __syncthreads();

    if (lane < 16) {
        const int srcLane = (m < 8) ? m : (m + 16);
        const int r       = (m < 8) ? m : (m - 8);
        out[row] = sc[srcLane * 8 + r];
    }
}

// ---------------------------------------------------------------------------
// inputs: d_in[0]=x (8192x6 f32), d_in[1]=decision_embed (unused), d_in[2]=b (4096 f32)
// output: 8192 f32. d_ws: 4096-float lattice scratch.
// ---------------------------------------------------------------------------
extern "C" void kernel_launch(void* const* d_in, const int* in_sizes, int n_in,
                              void* d_out, int out_size, void* d_ws, size_t ws_size,
                              hipStream_t stream) {
    const float* x = (const float*)d_in[0];
    const float* b = (const float*)d_in[2];
    float* lat = (float*)d_ws;
    float* out = (float*)d_out;

    hl_lattice_kernel<<<1, 1024, 0, stream>>>(b, lat);
    hl_interp_kernel<<<64, 256, 0, stream>>>(x, lat, out);
}